// SphericalHarmonics_69561290326720
// MI455X (gfx1250) — compile-verified
//
#include <hip/hip_runtime.h>
#include <stdint.h>

#define SH_PI 3.14159265358979323846
#define SH_TPB 128     // threads (=points) per block; 4 waves
#define SH_NF  98      // floats per point: 49 (l,m) slots * (re,im)

// ---------------------------------------------------------------------------
// Per-point spherical-harmonics recurrence (l = 0..6), exploiting the exact
// FP conjugate symmetry Y_{l,-m} = (-1)^m conj(Y_{l,m}) of the reference
// recurrence: only m >= 0 is computed (7-entry arrays), negative m emitted as
// sign-flipped copies. Fully unrolled: every coefficient sqrt folds to an f32
// literal and every array index is constant (arrays live in VGPRs).
// Writes 49 float2 (re,im) pairs to dst.
// ---------------------------------------------------------------------------
__device__ __forceinline__ void sh_compute(float x, float y, float z,
                                           float2* __restrict__ dst) {
  const float n2  = x * x + y * y + z * z;
  const float inv = (n2 > 0.0f) ? (1.0f / __builtin_sqrtf(n2)) : 0.0f;
  x *= inv; y *= inv; z *= inv;

  const float c1  = (float)__builtin_sqrt(3.0 / (4.0 * SH_PI));
  const float iv2 = (float)__builtin_sqrt(0.5);

  // l = 0
  dst[0] = make_float2((float)__builtin_sqrt(1.0 / (4.0 * SH_PI)), 0.0f);

  // l = 1: keep m >= 0 only; m = -1 is the exact mirror.
  float Rp[7], Ip[7];
  Rp[0] =  z * c1;        Ip[0] = 0.0f;
  Rp[1] = -x * iv2 * c1;  Ip[1] = -y * iv2 * c1;

  const float ppr =  Rp[1], ppi = Ip[1];   // p1 at m=+1
  const float pmr = -Rp[1], pmi = Ip[1];   // p1 at m=-1 (exact mirror)
  const float p0r =  Rp[0];                // p1 at m=0 (imag part is exactly 0)

  dst[1] = make_float2(pmr,  pmi);         // m=-1
  dst[2] = make_float2(Rp[0], 0.0f);       // m= 0
  dst[3] = make_float2(Rp[1], Ip[1]);      // m=+1

  #pragma unroll
  for (int l = 2; l <= 6; ++l) {
    const float scale = (float)__builtin_sqrt(4.0 * SH_PI * (double)(2 * l + 1)
                                              / (3.0 * (double)l));
    float Rc[7], Ic[7];
    #pragma unroll
    for (int m = 0; m <= l; ++m) {
      const float Cp = (float)__builtin_sqrt(
          (double)((l + m - 1) * (l + m)) / (double)((2 * l - 1) * (2 * l)));
      const float C0 = (float)__builtin_sqrt(
          (double)((l - m) * (l + m)) / (double)((2 * l - 1) * l));
      const float Cm = (float)__builtin_sqrt(
          (double)((l - m - 1) * (l - m)) / (double)((2 * l - 1) * (2 * l)));

      // prev level holds m' = 0..l-1; prev[-1] is the exact mirror of prev[1].
      const float aR = (m == 0) ? -Rp[1] : Rp[m - 1];
      const float aI = (m == 0) ?  Ip[1] : Ip[m - 1];
      const float bR = (m <= l - 1) ? Rp[m]     : 0.0f;
      const float bI = (m <= l - 1) ? Ip[m]     : 0.0f;
      const float cR = (m <= l - 2) ? Rp[m + 1] : 0.0f;
      const float cI = (m <= l - 2) ? Ip[m + 1] : 0.0f;

      // p0 imag == 0 exactly -> those products dropped (compiler can't).
      const float re = scale * (Cp * (aR * ppr - aI * ppi) +
                                C0 * (bR * p0r) +
                                Cm * (cR * pmr - cI * pmi));
      const float im = (m == 0) ? 0.0f   // exact cancellation (Cp==Cm, Ip[0]==0)
                     : scale * (Cp * (aR * ppi + aI * ppr) +
                                C0 * (bI * p0r) +
                                Cm * (cR * pmi + cI * pmr));
      Rc[m] = re; Ic[m] = im;
      dst[l * l + l + m] = make_float2(re, im);
      if (m > 0) {                       // Y_{l,-m} = (-1)^m conj(Y_{l,m})
        if (m & 1) dst[l * l + l - m] = make_float2(-re,  im);
        else       dst[l * l + l - m] = make_float2( re, -im);
      }
    }
    #pragma unroll
    for (int m = 0; m <= l; ++m) { Rp[m] = Rc[m]; Ip[m] = Ic[m]; }
  }
}

// ---------------------------------------------------------------------------
// Main kernel: one point per thread. Results staged in LDS (row stride 98
// floats = 49 float2 -> gcd(49,32)=1 => conflict-free ds_store_b64), then each
// wave drains its own contiguous 32-point slice to HBM with the gfx1250 async
// LDS->global store engine (16B per lane, fully coalesced, ASYNCcnt-tracked,
// non-temporal: output is write-once). No barrier: waves touch only their own
// LDS slice; s_wait_dscnt orders our DS writes against the async engine.
// ---------------------------------------------------------------------------
__global__ __launch_bounds__(SH_TPB) void sh_main_kernel(
    const float* __restrict__ pos, float* __restrict__ out) {
  __shared__ __align__(16) float sh[SH_TPB * SH_NF];  // 49 KB; no padding so
                                                      // LDS-linear == global-linear
  const unsigned tid = threadIdx.x;
  const unsigned g   = blockIdx.x * SH_TPB + tid;

  const float x = pos[3u * g + 0u];
  const float y = pos[3u * g + 1u];
  const float z = pos[3u * g + 2u];

  sh_compute(x, y, z, (float2*)(sh + tid * SH_NF));

  const unsigned lane = tid & 31u;
  const unsigned wave = tid >> 5;

  // DScnt and ASYNCcnt are independent counters: drain our DS writes first.
  asm volatile("s_wait_dscnt 0x0" ::: "memory");

  // Wave-local contiguous copy: 32 points * 98 floats = 784 * 16B chunks.
  const unsigned ldsBase = (unsigned)(uintptr_t)sh + wave * (32u * SH_NF * 4u);
  const unsigned gBase   = (blockIdx.x * SH_TPB + wave * 32u) * (SH_NF * 4u);
  #pragma unroll
  for (unsigned k = 0; k < 25u; ++k) {
    const unsigned idx = lane + 32u * k;
    if (idx < (32u * SH_NF / 4u)) {  // 784; only k==24 is partially active
      asm volatile("global_store_async_from_lds_b128 %0, %1, %2 th:TH_STORE_NT"
                   :: "v"(gBase + idx * 16u),   // 32-bit byte offset (<2^31)
                      "v"(ldsBase + idx * 16u), // LDS byte address
                      "s"(out)                  // 64-bit SGPR base
                   : "memory");
    }
  }
  asm volatile("s_wait_asynccnt 0x0" ::: "memory");
}

// Tail kernel for N % SH_TPB != 0 (not hit for N = 2^20, kept for generality).
__global__ void sh_tail_kernel(const float* __restrict__ pos,
                               float* __restrict__ out, int start, int N) {
  const int g = start + (int)(blockIdx.x * blockDim.x + threadIdx.x);
  if (g >= N) return;
  sh_compute(pos[3 * g], pos[3 * g + 1], pos[3 * g + 2],
             (float2*)(out + (size_t)g * SH_NF));
}

extern "C" void kernel_launch(void* const* d_in, const int* in_sizes, int n_in,
                              void* d_out, int out_size, void* d_ws, size_t ws_size,
                              hipStream_t stream) {
  const float* pos = (const float*)d_in[0];
  float*       out = (float*)d_out;
  const int N  = in_sizes[0] / 3;
  const int nb = N / SH_TPB;
  if (nb > 0) {
    sh_main_kernel<<<nb, SH_TPB, 0, stream>>>(pos, out);
  }
  const int rem = N - nb * SH_TPB;
  if (rem > 0) {
    sh_tail_kernel<<<(rem + 63) / 64, 64, 0, stream>>>(pos, out, nb * SH_TPB, N);
  }
}